// binnings_25864293056764
// MI455X (gfx1250) — compile-verified
//
#include <hip/hip_runtime.h>

// out[b,g,h,w] = sum_n w[g,n] * x[b, (g/4)*16 + 4n + g%4, h, w]
// x: (8,64,512,512) f32, w: (16,4) f32, out: (8,16,512,512) f32.
// Pure streaming, ~670 MB moved, AI ~0.4 flop/B -> HBM-bound (~29us floor @23.3TB/s).

typedef float f32x4 __attribute__((ext_vector_type(4)));
typedef int   i32x4 __attribute__((ext_vector_type(4)));

#define PLANE (512 * 512)   // floats per channel plane
#define TPB 256             // 8 wave32s per workgroup
#define GRIDX 32            // chunks strided across 32 blocks per plane

// ---- CDNA5 async-to-LDS path detection (device pass only) ----------------
#if defined(__gfx1250__) && __has_builtin(__builtin_amdgcn_global_load_async_to_lds_b128)
#define DEVICE_HAS_ASYNC 1
#else
#define DEVICE_HAS_ASYNC 0
#endif

#if DEVICE_HAS_ASYNC
// builtin signature (from hipcc diagnostic): param0 = AS1 int4*, so use
// explicit address-space-qualified int4 pointers for both sides.
typedef __attribute__((address_space(1))) i32x4* g_i32x4_ptr;
typedef __attribute__((address_space(3))) i32x4* l_i32x4_ptr;
#if __has_builtin(__builtin_amdgcn_s_wait_asynccnt)
#define ASYNC_WAIT(n) __builtin_amdgcn_s_wait_asynccnt(n)
#else
#define ASYNC_WAIT(n) asm volatile("s_wait_asynccnt %0" ::"i"(n) : "memory")
#endif
#endif

__global__ __launch_bounds__(TPB) void binnings_weighted_gather(
    const float* __restrict__ x, const float* __restrict__ gw,
    float* __restrict__ out) {
  const int plane = blockIdx.y;              // b*16 + g, 0..127
  const int g = plane & 15;
  const int b = plane >> 4;
  const int cbase = ((g >> 2) << 4) + (g & 3);
  const float* xp = x + (size_t)(b * 64 + cbase) * PLANE;
  float* op = out + (size_t)plane * PLANE;
  // uniform per block -> scalar loads
  const float w0 = gw[4 * g + 0];
  const float w1 = gw[4 * g + 1];
  const float w2 = gw[4 * g + 2];
  const float w3 = gw[4 * g + 3];
  const int tid = threadIdx.x;
  const int nchunks = PLANE / (4 * TPB);     // 256 chunks of 1024 px per plane
  const int cstride = gridDim.x;

#if DEVICE_HAS_ASYNC
  // Double-buffered DMA staging: 2 bufs * 4 ch * 256 lanes * 16B = 32KB LDS.
  // Lane i reads back only the slots lane i loaded, so the per-wave
  // ASYNCcnt wait is the only synchronization required (no barriers).
  __shared__ f32x4 lbuf[2][4][TPB];
  auto issue = [&](int buf, int c) {
    const f32x4* src = (const f32x4*)xp + (size_t)c * TPB + tid;
#pragma unroll
    for (int ch = 0; ch < 4; ++ch) {
      __builtin_amdgcn_global_load_async_to_lds_b128(
          (g_i32x4_ptr)(src + (size_t)ch * PLANE),
          (l_i32x4_ptr)&lbuf[buf][ch][tid],
          /*offset=*/0, /*cpol=*/0);
    }
  };

  int c = blockIdx.x;                        // GRIDX < nchunks always
  int buf = 0;
  issue(buf, c);                             // prime the pipeline (4 async ops)
  for (;;) {
    const int cn = c + cstride;
    const bool more = (cn < nchunks);
    if (more) {
      issue(buf ^ 1, cn);                    // next chunk in flight (8 total)
      ASYNC_WAIT(4);                         // wait only for current chunk
    } else {
      ASYNC_WAIT(0);
    }
    asm volatile("" ::: "memory");           // keep LDS reads below the wait
    f32x4 v0 = lbuf[buf][0][tid];
    f32x4 v1 = lbuf[buf][1][tid];
    f32x4 v2 = lbuf[buf][2][tid];
    f32x4 v3 = lbuf[buf][3][tid];
    f32x4 r = v0 * w0 + v1 * w1 + v2 * w2 + v3 * w3;
    __builtin_nontemporal_store(r, (f32x4*)op + (size_t)c * TPB + tid);
    if (!more) break;
    c = cn;
    buf ^= 1;
  }
#else
  // Direct-streaming fallback: b128 NT loads, 16 FMAs, b128 NT store.
  for (int c = blockIdx.x; c < nchunks; c += cstride) {
    const f32x4* src = (const f32x4*)xp + (size_t)c * TPB + tid;
    f32x4 v0 = __builtin_nontemporal_load(src);
    f32x4 v1 = __builtin_nontemporal_load(src + (size_t)PLANE);
    f32x4 v2 = __builtin_nontemporal_load(src + (size_t)2 * PLANE);
    f32x4 v3 = __builtin_nontemporal_load(src + (size_t)3 * PLANE);
    f32x4 r = v0 * w0 + v1 * w1 + v2 * w2 + v3 * w3;
    __builtin_nontemporal_store(r, (f32x4*)op + (size_t)c * TPB + tid);
  }
#endif
}

extern "C" void kernel_launch(void* const* d_in, const int* in_sizes, int n_in,
                              void* d_out, int out_size, void* d_ws,
                              size_t ws_size, hipStream_t stream) {
  (void)in_sizes; (void)n_in; (void)d_ws; (void)ws_size; (void)out_size;
  const float* x = (const float*)d_in[0];    // (8,64,512,512) f32
  const float* gw = (const float*)d_in[1];   // (16,4,1,1) f32
  float* out = (float*)d_out;                // (8,16,512,512) f32
  dim3 grid(GRIDX, 128, 1);                  // 4096 blocks * 8 waves
  binnings_weighted_gather<<<grid, TPB, 0, stream>>>(x, gw, out);
}